// SGP_LAYER_77670188581411
// MI455X (gfx1250) — compile-verified
//
#include <hip/hip_runtime.h>
#include <hip/hip_bf16.h>

// Problem constants (from reference)
#define H_    12
#define L_    1024
#define HDIM_ 768
#define D_    64
#define NBH   96      // B*H batches
#define LAMC  0.01f
#define EPSC  0.002f

typedef __attribute__((ext_vector_type(2))) float v2f;
typedef __attribute__((ext_vector_type(8))) float v8f;

enum { EPI_NONE = 0, EPI_BIAS = 1, EPI_RFF = 2, EPI_U = 3 };
enum { OFF_DENSE = 0, OFF_HEAD = 1, OFF_SHARED = 2 };

__device__ __forceinline__ long boff(int mode, int bh, long stride) {
  if (mode == OFF_DENSE) return (long)bh * stride;
  if (mode == OFF_HEAD) { int b = bh / H_; int h = bh - b * H_;
                          return (long)b * (long)L_ * HDIM_ + (long)h * D_; }
  return 0;
}

#define WMMA4(a, b, c) \
  __builtin_amdgcn_wmma_f32_16x16x4_f32(false, (a), false, (b), (short)0, (c), false, false)

// ---------------------------------------------------------------------------
// Batched GEMM on V_WMMA_F32_16X16X4_F32.
// C[bh] (M x N) = op(A[bh]) * op(B[bh]), fused epilogue.
// Block 256 threads = 8 wave32, 128x64 C tile; each wave owns 32x32
// (4 fp32 accumulators -> 16 WMMA per K-stage, fragments amortized 2x).
// Stage-in is b128 global loads + b128 LDS stores, register double-buffered
// so the next HBM tile is in flight while WMMAs drain the current LDS tile.
// LDS row strides 20 / 68: bank-conflict-free and 16B-aligned.
// Transpose flags are template params -> branchless loaders.
// ---------------------------------------------------------------------------
template<int EPI, int TA, int TB>
__global__ __launch_bounds__(256) void gemm_wmma(
    const float* __restrict__ A, const float* __restrict__ Bm,
    float* __restrict__ Cm, const float* __restrict__ aux,
    int Kdim, int lda, int ldb, int ldc,
    long sA, long sB, long sC, int mA, int mB, int mC)
{
  constexpr int BM = 128, BN = 64, BK = 16, LAS = 20, LBS = 68;
  __shared__ __align__(16) float As[BM * LAS];   // 10.0 KB
  __shared__ __align__(16) float Bs[BK * LBS];   //  4.25 KB

  const int  bh    = blockIdx.z;
  const long baseA = boff(mA, bh, sA);
  const long baseB = boff(mB, bh, sB);
  const long baseC = boff(mC, bh, sC);
  const int  tileM = blockIdx.y * BM, tileN = blockIdx.x * BN;
  const int  tid = threadIdx.x, lane = tid & 31, wave = tid >> 5;
  const int  wm = wave & 3, wn = wave >> 2;

  v8f acc00 = {0.f,0.f,0.f,0.f,0.f,0.f,0.f,0.f};
  v8f acc01 = acc00, acc10 = acc00, acc11 = acc00;

  float4 ra0, ra1, rb;
  auto fetchA = [&](int kt, float4& r0, float4& r1) {
    if (TA == 0) {                       // contiguous along K
      int m0 = tid >> 2,          k0 = (tid & 3) << 2;
      int m1 = (tid + 256) >> 2,  k1 = (tid & 3) << 2;
      r0 = *(const float4*)&A[baseA + (long)(tileM + m0) * lda + (kt + k0)];
      r1 = *(const float4*)&A[baseA + (long)(tileM + m1) * lda + (kt + k1)];
    } else {                             // contiguous along M
      int k0 = tid >> 5,         mq0 = (tid & 31) << 2;
      int k1 = (tid + 256) >> 5, mq1 = (tid & 31) << 2;
      r0 = *(const float4*)&A[baseA + (long)(kt + k0) * lda + (tileM + mq0)];
      r1 = *(const float4*)&A[baseA + (long)(kt + k1) * lda + (tileM + mq1 + 64)];
    }
  };
  auto storeA = [&](const float4& r0, const float4& r1) {
    if (TA == 0) {
      int m0 = tid >> 2,         k0 = (tid & 3) << 2;
      int m1 = (tid + 256) >> 2;
      *(float4*)&As[m0 * LAS + k0] = r0;
      *(float4*)&As[m1 * LAS + k0] = r1;
    } else {
      int k0 = tid >> 5,         mq0 = (tid & 31) << 2;
      int k1 = (tid + 256) >> 5, mq1 = ((tid & 31) << 2) + 64;
      As[(mq0+0)*LAS + k0] = r0.x; As[(mq0+1)*LAS + k0] = r0.y;
      As[(mq0+2)*LAS + k0] = r0.z; As[(mq0+3)*LAS + k0] = r0.w;
      As[(mq1+0)*LAS + k1] = r1.x; As[(mq1+1)*LAS + k1] = r1.y;
      As[(mq1+2)*LAS + k1] = r1.z; As[(mq1+3)*LAS + k1] = r1.w;
    }
  };
  auto fetchB = [&](int kt, float4& r) {
    if (TB == 0) { int k = tid >> 4, n4 = (tid & 15) << 2;
      r = *(const float4*)&Bm[baseB + (long)(kt + k) * ldb + (tileN + n4)];
    } else {       int n = tid >> 2, kq = (tid & 3) << 2;
      r = *(const float4*)&Bm[baseB + (long)(tileN + n) * ldb + (kt + kq)];
    }
  };
  auto storeB = [&](const float4& r) {
    if (TB == 0) { int k = tid >> 4, n4 = (tid & 15) << 2;
      *(float4*)&Bs[k * LBS + n4] = r;
    } else {       int n = tid >> 2, kq = (tid & 3) << 2;
      Bs[(kq+0)*LBS + n] = r.x; Bs[(kq+1)*LBS + n] = r.y;
      Bs[(kq+2)*LBS + n] = r.z; Bs[(kq+3)*LBS + n] = r.w;
    }
  };

  fetchA(0, ra0, ra1);
  fetchB(0, rb);

  const int mrow0 = (wm * 32 + (lane & 15)) * LAS;
  const int mrow1 = mrow0 + 16 * LAS;
  const int kh    = (lane >> 4) << 1;          // lanes 16..31 carry K=+2,+3
  const int nc0   = wn * 32 + (lane & 15), nc1 = nc0 + 16;

  for (int kt = 0; kt < Kdim; kt += BK) {
    storeA(ra0, ra1);
    storeB(rb);
    __syncthreads();
    if (kt + BK < Kdim) {                      // overlap next HBM tile w/ WMMA
      fetchA(kt + BK, ra0, ra1);
      fetchB(kt + BK, rb);
    }
#pragma unroll
    for (int kk = 0; kk < BK; kk += 4) {
      v2f a0, a1, b0, b1;
      a0.x = As[mrow0 + kk + kh];  a0.y = As[mrow0 + kk + kh + 1];
      a1.x = As[mrow1 + kk + kh];  a1.y = As[mrow1 + kk + kh + 1];
      b0.x = Bs[(kk + kh) * LBS + nc0]; b0.y = Bs[(kk + kh + 1) * LBS + nc0];
      b1.x = Bs[(kk + kh) * LBS + nc1]; b1.y = Bs[(kk + kh + 1) * LBS + nc1];
      acc00 = WMMA4(a0, b0, acc00);  acc01 = WMMA4(a0, b1, acc01);
      acc10 = WMMA4(a1, b0, acc10);  acc11 = WMMA4(a1, b1, acc11);
    }
    __syncthreads();
  }

  // Epilogue. 16x16 frag: VGPR j -> row base + (lane>>4)*8 + j, col = lane&15.
  const int r0 = tileM + wm * 32 + ((lane >> 4) << 3);
  const int c0 = tileN + wn * 32 + (lane & 15);
  const long vb = (EPI == EPI_U) ? boff(OFF_HEAD, bh, 0) : 0;
#pragma unroll
  for (int j = 0; j < 8; j++) {
    float v[4] = { acc00[j], acc01[j], acc10[j], acc11[j] };
    int   rr[4] = { r0 + j, r0 + j, r0 + 16 + j, r0 + 16 + j };
    int   cc[4] = { c0, c0 + 16, c0, c0 + 16 };
#pragma unroll
    for (int t = 0; t < 4; t++) {
      float o = v[t];
      if (EPI == EPI_BIAS)      o += aux[cc[t]];
      else if (EPI == EPI_RFF)  o = 0.08838834764831845f * cosf(o + aux[cc[t]]);
      else if (EPI == EPI_U)    o = (aux[vb + (long)rr[t] * HDIM_ + cc[t]] - o) * 100.0f;
      Cm[baseC + (long)rr[t] * ldc + cc[t]] = o;
    }
  }
}

// ---------------------------------------------------------------------------
// Batched Cholesky of (coefA*A + coefC*C + jitter*I): one 256x256 fp32 matrix
// per workgroup, resident in CDNA5's 320KB LDS (row stride 257 -> conflict-
// free column ops). Emits L and 2*sum(log diag L). Cin must be valid (pass
// Ain with coefC=0 when unused) -> branchless init.
// ---------------------------------------------------------------------------
__global__ __launch_bounds__(256) void chol_kernel(
    const float* __restrict__ Ain, const float* __restrict__ Cin,
    float coefA, float coefC, float jitter,
    float* __restrict__ Lout, float* __restrict__ logdet2)
{
  extern __shared__ float sm[];                  // 256*257 floats
  __shared__ float sdiag;
  __shared__ float red[256];
  const int LDP = 257;
  const int bh = blockIdx.x, tid = threadIdx.x;
  const long base = (long)bh * 65536;

  for (int idx = tid; idx < 65536; idx += 256) {
    int r = idx >> 8, c = idx & 255;
    float v = coefA * Ain[base + idx] + coefC * Cin[base + idx];
    if (r == c) v += jitter;
    sm[r * LDP + c] = v;
  }
  __syncthreads();

  for (int k = 0; k < 256; k++) {
    if (tid == 0) { float d = sqrtf(sm[k * LDP + k]); sm[k * LDP + k] = d; sdiag = 1.0f / d; }
    __syncthreads();
    const float inv = sdiag;
    const int i = tid;
    float lik = 0.f;
    if (i > k) { lik = sm[i * LDP + k] * inv; sm[i * LDP + k] = lik; }
    __syncthreads();
    if (i > k) {
      for (int j = k + 1; j <= i; j++) sm[i * LDP + j] -= lik * sm[j * LDP + k];
    }
    __syncthreads();
  }

  for (int idx = tid; idx < 65536; idx += 256) {
    int r = idx >> 8, c = idx & 255;
    Lout[base + idx] = (c <= r) ? sm[r * LDP + c] : 0.f;
  }
  red[tid] = logf(sm[tid * LDP + tid]);
  __syncthreads();
  for (int s = 128; s > 0; s >>= 1) { if (tid < s) red[tid] += red[tid + s]; __syncthreads(); }
  if (tid == 0) logdet2[bh] = 2.0f * red[0];
}

// ---------------------------------------------------------------------------
// Batched cholesky_solve: X = (L L^T)^{-1} R. One thread per RHS column,
// columns fully independent -> zero barriers; panel lives in LDS.
// NRHS compile-time -> shift addressing.
// ---------------------------------------------------------------------------
template<int NRHS>
__global__ void cho_solve_kernel(const float* __restrict__ Lm,
                                 const float* __restrict__ Rin,
                                 float* __restrict__ Xout)
{
  extern __shared__ float xs[];                  // 256*NRHS floats
  const int col = threadIdx.x, bh = blockIdx.x;
  const float* Lb = Lm + (long)bh * 65536;
  const float* Rb = Rin + (long)bh * 256 * NRHS;
  float* Xb = Xout + (long)bh * 256 * NRHS;

  for (int k = 0; k < 256; k++) xs[k * NRHS + col] = Rb[k * NRHS + col];
  for (int k = 0; k < 256; k++) {                // L y = r
    float s = xs[k * NRHS + col];
    for (int j = 0; j < k; j++) s -= Lb[k * 256 + j] * xs[j * NRHS + col];
    xs[k * NRHS + col] = s / Lb[k * 257];
  }
  for (int k = 255; k >= 0; k--) {               // L^T x = y
    float s = xs[k * NRHS + col];
    for (int j = k + 1; j < 256; j++) s -= Lb[j * 256 + k] * xs[j * NRHS + col];
    xs[k * NRHS + col] = s / Lb[k * 257];
  }
  for (int k = 0; k < 256; k++) Xb[k * NRHS + col] = xs[k * NRHS + col];
}

// out[bh] = sum X .* (c1*Y + c2*Z). Z must be valid (pass Y with c2=0).
__global__ __launch_bounds__(256) void dot_combine(
    const float* __restrict__ X, const float* __restrict__ Y,
    const float* __restrict__ Z, float c1, float c2,
    float* __restrict__ out, int per)
{
  __shared__ float red[256];
  const int bh = blockIdx.x, tid = threadIdx.x;
  const long base = (long)bh * per;
  float acc = 0.f;
  for (int i = tid; i < per; i += 256)
    acc += X[base + i] * (c1 * Y[base + i] + c2 * Z[base + i]);
  red[tid] = acc; __syncthreads();
  for (int s = 128; s > 0; s >>= 1) { if (tid < s) red[tid] += red[tid + s]; __syncthreads(); }
  if (tid == 0) out[bh] = red[0];
}

// yy[bh] = sum over head-view of v^2
__global__ __launch_bounds__(256) void yy_kernel(const float* __restrict__ Pv,
                                                 float* __restrict__ out)
{
  __shared__ float red[256];
  const int bh = blockIdx.x, tid = threadIdx.x;
  const int b = bh / H_, h = bh - b * H_;
  const float* basep = Pv + (long)b * L_ * HDIM_ + (long)h * D_;
  float acc = 0.f;
  for (int idx = tid; idx < L_ * D_; idx += 256) {
    int n = idx >> 6, d = idx & 63;
    float v = basep[(long)n * HDIM_ + d];
    acc += v * v;
  }
  red[tid] = acc; __syncthreads();
  for (int s = 128; s > 0; s >>= 1) { if (tid < s) red[tid] += red[tid + s]; __syncthreads(); }
  if (tid == 0) out[bh] = red[0];
}

// CGP lower bound. logdet identity: A + (lam/eps)C = lam*S, so
// logdet(I + (lam/eps) A^{-1} C) = M log lam + 2*sumlog(diag L_S) - 2*sumlog(diag Lo)
__global__ __launch_bounds__(256) void final_kernel(
    const float* __restrict__ Gk, const float* __restrict__ yy,
    const float* __restrict__ df2, const float* __restrict__ modq,
    const float* __restrict__ modk, const float* __restrict__ trpart,
    const float* __restrict__ ldLo, const float* __restrict__ ldLSq,
    const float* __restrict__ ldLSk, float* __restrict__ reg_out)
{
  __shared__ float red[256];
  const int tid = threadIdx.x;
  const float Mf = 256.f, Nf = 1024.f, Df = 64.f;
  const float le = logf(EPSC), ll = logf(LAMC), l2pi = logf(6.2831853071795864f);
  float acc = 0.f;
  for (int bh = tid; bh < NBH; bh += 256) {
    float tr = 0.f;
    const float* g = Gk + (long)bh * 65536;
    for (int i = 0; i < 256; i++) tr += g[i * 257];
    float datafit_q = yy[bh] / EPSC - df2[bh] / (EPSC * EPSC);
    float logdet_Iq = Mf * ll + ldLSq[bh] - ldLo[bh];
    float LBq = -0.5f * (datafit_q + Df * modq[bh] + Nf * le + logdet_Iq + Nf * l2pi);
    float datafit_k = Df * (tr / EPSC - trpart[bh] / (EPSC * EPSC));
    float logdet_Ik = Mf * ll + ldLSk[bh] - ldLo[bh];
    float LBk = -0.5f * (datafit_k + Df * modk[bh] + Nf * le + logdet_Ik + Nf * l2pi);
    acc += LBq + LBk;
  }
  red[tid] = acc; __syncthreads();
  for (int s = 128; s > 0; s >>= 1) { if (tid < s) red[tid] += red[tid + s]; __syncthreads(); }
  if (tid == 0) reg_out[0] = -red[0] / (float)NBH;
}

// ---------------------------------------------------------------------------
template<int EPI, int TA, int TB>
static void gemmT(hipStream_t s, const float* A, const float* B, float* C,
                  const float* aux, int M, int N, int K,
                  int lda, int ldb, int ldc,
                  long sA, long sB, long sC, int mA, int mB, int mC, int nb)
{
  dim3 g(N / 64, M / 128, nb);
  gemm_wmma<EPI, TA, TB><<<g, dim3(256), 0, s>>>(A, B, C, aux, K, lda, ldb, ldc,
                                                 sA, sB, sC, mA, mB, mC);
}

extern "C" void kernel_launch(void* const* d_in, const int* in_sizes, int n_in,
                              void* d_out, int out_size, void* d_ws, size_t ws_size,
                              hipStream_t stream) {
  (void)in_sizes; (void)n_in; (void)out_size; (void)ws_size;
  const float* x    = (const float*)d_in[0];
  const float* Wq   = (const float*)d_in[1];  const float* bq  = (const float*)d_in[2];
  const float* Wk   = (const float*)d_in[3];  const float* bk  = (const float*)d_in[4];
  const float* Wv   = (const float*)d_in[5];  const float* bv  = (const float*)d_in[6];
  const float* Wx0  = (const float*)d_in[7];  const float* bx0 = (const float*)d_in[8];
  const float* Wo   = (const float*)d_in[9];  const float* bo  = (const float*)d_in[10];
  const float* omega= (const float*)d_in[11]; const float* brff= (const float*)d_in[12];
  float* out = (float*)d_out;                 // [8*1024*768] forward + [1] reg

  // ---- workspace layout (floats) ----
  float* w = (float*)d_ws; size_t o = 0;
  auto alloc = [&](size_t n) { float* p = w + o; o += n; return p; };
  const size_t PRJ = 8192ul * 768, PHI = 96ul * 1024 * 256, MMB = 96ul * 65536,
               SB = 96ul * 256 * 64;
  float *Pq = alloc(PRJ), *Pk = alloc(PRJ), *Pv = alloc(PRJ), *Px0 = alloc(PRJ);
  float *PHIq = alloc(PHI), *PHIk = alloc(PHI), *PHIo = alloc(PHI);
  float *Gk = alloc(MMB), *Go = alloc(MMB), *Cq = alloc(MMB);
  float *Lk = alloc(MMB), *Lo = alloc(MMB), *LSq = alloc(MMB), *LSk = alloc(MMB);
  float *GA = alloc(MMB), *TT = alloc(MMB), *PP = alloc(MMB);
  float *U  = alloc(MMB), *Ybuf = alloc(PRJ);
  float *SB1 = alloc(SB), *SB2 = alloc(SB), *SB3 = alloc(SB), *SB4 = alloc(SB), *SB5 = alloc(SB);
  float *ldLk = alloc(96), *ldLo = alloc(96), *ldLSq = alloc(96), *ldLSk = alloc(96);
  float *df2 = alloc(96), *modq = alloc(96), *modk = alloc(96), *trp = alloc(96), *yyb = alloc(96);
  (void)ldLk;

  const int DN = OFF_DENSE, HD = OFF_HEAD, SH = OFF_SHARED;
  const size_t CHOL_LDS = 256 * 257 * sizeof(float);

  // 1) projections: P = x @ W^T + b
  gemmT<EPI_BIAS,0,1>(stream, x, Wq,  Pq,  bq,  8192, 768, 768, 768, 768, 768, 0,0,0, SH,SH,SH, 1);
  gemmT<EPI_BIAS,0,1>(stream, x, Wk,  Pk,  bk,  8192, 768, 768, 768, 768, 768, 0,0,0, SH,SH,SH, 1);
  gemmT<EPI_BIAS,0,1>(stream, x, Wv,  Pv,  bv,  8192, 768, 768, 768, 768, 768, 0,0,0, SH,SH,SH, 1);
  gemmT<EPI_BIAS,0,1>(stream, x, Wx0, Px0, bx0, 8192, 768, 768, 768, 768, 768, 0,0,0, SH,SH,SH, 1);

  // 2) RFF: Phi = sqrt(2/M) cos(head @ omega + b_rff), per (b,h)
  gemmT<EPI_RFF,0,0>(stream, Pq,  omega, PHIq, brff, 1024, 256, 64, 768, 256, 256, 0,0,262144, HD,SH,DN, NBH);
  gemmT<EPI_RFF,0,0>(stream, Pk,  omega, PHIk, brff, 1024, 256, 64, 768, 256, 256, 0,0,262144, HD,SH,DN, NBH);
  gemmT<EPI_RFF,0,0>(stream, Px0, omega, PHIo, brff, 1024, 256, 64, 768, 256, 256, 0,0,262144, HD,SH,DN, NBH);

  // 3) Gram matrices Phi^T Phi
  gemmT<EPI_NONE,1,0>(stream, PHIk, PHIk, Gk, nullptr, 256, 256, 1024, 256, 256, 256, 262144,262144,65536, DN,DN,DN, NBH);
  gemmT<EPI_NONE,1,0>(stream, PHIo, PHIo, Go, nullptr, 256, 256, 1024, 256, 256, 256, 262144,262144,65536, DN,DN,DN, NBH);
  gemmT<EPI_NONE,1,0>(stream, PHIq, PHIq, Cq, nullptr, 256, 256, 1024, 256, 256, 256, 262144,262144,65536, DN,DN,DN, NBH);

  // 4) Phi_k^T v
  gemmT<EPI_NONE,1,0>(stream, PHIk, Pv, SB1, nullptr, 256, 64, 1024, 256, 768, 64, 262144,0,16384, DN,HD,DN, NBH);

  // 5) Cholesky: Lk=chol(Gk+lam I), Lo=chol(Go+lam I),
  //    L_Sq=chol(Go/lam + I + Cq/eps), L_Sk=chol(Go/lam + I + Gk/eps)
  chol_kernel<<<NBH, 256, CHOL_LDS, stream>>>(Gk, Gk, 1.f, 0.f, LAMC, Lk, ldLk);
  chol_kernel<<<NBH, 256, CHOL_LDS, stream>>>(Go, Go, 1.f, 0.f, LAMC, Lo, ldLo);
  chol_kernel<<<NBH, 256, CHOL_LDS, stream>>>(Go, Cq, 100.f, 500.f, 1.f, LSq, ldLSq);
  chol_kernel<<<NBH, 256, CHOL_LDS, stream>>>(Go, Gk, 100.f, 500.f, 1.f, LSk, ldLSk);

  // 6) forward path
  cho_solve_kernel<64><<<NBH, 64, 256*64*4, stream>>>(Lk, SB1, SB1);                // Z1
  gemmT<EPI_U,0,0>(stream, PHIk, SB1, U, Pv, 1024, 64, 256, 256, 64, 64, 262144,16384,65536, DN,DN,DN, NBH);
  gemmT<EPI_NONE,1,0>(stream, PHIk, U, SB2, nullptr, 256, 64, 1024, 256, 64, 64, 262144,65536,16384, DN,DN,DN, NBH);
  cho_solve_kernel<64><<<NBH, 64, 256*64*4, stream>>>(Lo, SB2, SB2);                // Ao^-1 Phi_k^T u
  gemmT<EPI_NONE,0,0>(stream, Go, SB2, SB3, nullptr, 256, 64, 256, 256, 64, 64, 65536,16384,16384, DN,DN,DN, NBH); // mid
  gemmT<EPI_NONE,0,0>(stream, PHIq, SB3, Ybuf, nullptr, 1024, 64, 256, 256, 64, 768, 262144,16384,0, DN,DN,HD, NBH); // y
  gemmT<EPI_BIAS,0,1>(stream, Ybuf, Wo, out, bo, 8192, 768, 768, 768, 768, 768, 0,0,0, SH,SH,SH, 1);

  // 7) regularizer pieces
  gemmT<EPI_NONE,1,0>(stream, PHIq, Pv, SB4, nullptr, 256, 64, 1024, 256, 768, 64, 262144,0,16384, DN,HD,DN, NBH); // PhiT_y
  cho_solve_kernel<64><<<NBH, 64, 256*64*4, stream>>>(LSq, SB4, SB5);
  dot_combine<<<NBH, 256, 0, stream>>>(SB4, SB5, SB5, 1.f, 0.f, df2, 16384);

  cho_solve_kernel<256><<<NBH, 256, 256*256*4, stream>>>(Lo, Go, GA);               // A^-1 G
  cho_solve_kernel<256><<<NBH, 256, 256*256*4, stream>>>(LSq, Cq, TT);              // S_q^-1 Cq
  gemmT<EPI_NONE,0,0>(stream, Cq, TT, PP, nullptr, 256, 256, 256, 256, 256, 256, 65536,65536,65536, DN,DN,DN, NBH);
  dot_combine<<<NBH, 256, 0, stream>>>(GA, Cq, PP, 500.f, -250000.f, modq, 65536);  // <GA, M_q>

  cho_solve_kernel<256><<<NBH, 256, 256*256*4, stream>>>(LSk, Gk, TT);              // S_k^-1 Ck
  dot_combine<<<NBH, 256, 0, stream>>>(Gk, TT, TT, 1.f, 0.f, trp, 65536);           // tr(Ck S^-1 Ck)
  gemmT<EPI_NONE,0,0>(stream, Gk, TT, PP, nullptr, 256, 256, 256, 256, 256, 256, 65536,65536,65536, DN,DN,DN, NBH);
  dot_combine<<<NBH, 256, 0, stream>>>(GA, Gk, PP, 500.f, -250000.f, modk, 65536);  // <GA, M_k>

  yy_kernel<<<NBH, 256, 0, stream>>>(Pv, yyb);
  final_kernel<<<1, 256, 0, stream>>>(Gk, yyb, df2, modq, modk, trp, ldLo, ldLSq, ldLSk,
                                      out + 8192ul * 768);
}